// PMM_67216238182716
// MI455X (gfx1250) — compile-verified
//
#include <hip/hip_runtime.h>
#include <hip/hip_bf16.h>

#define BB   16
#define LL   128
#define SS   128
#define DD   512
#define EPSV 1e-12f
#define SHARP 2.0f

typedef float v2f __attribute__((ext_vector_type(2)));
typedef float v8f __attribute__((ext_vector_type(8)));

__device__ __forceinline__ float wave_sum(float v) {
    #pragma unroll
    for (int off = 16; off > 0; off >>= 1) v += __shfl_xor(v, off, 32);
    return v;
}
__device__ __forceinline__ float wave_max(float v) {
    #pragma unroll
    for (int off = 16; off > 0; off >>= 1) v = fmaxf(v, __shfl_xor(v, off, 32));
    return v;
}

// ---------------------------------------------------------------------------
// Kernel 1: batched projection GEMM  (2048 x 512) @ (512 x 512) + bias,
// for Wq / Wk / Wv selected by blockIdx.y.  Uses V_WMMA_F32_16X16X4_F32.
// A-frag (16x4 f32): lane<16 -> row=lane, K=k0,k0+1 ; lane>=16 -> K=k0+2,k0+3
// B-frag (4x16 f32): lane<16 -> K=k0,k0+1, col=lane ; lane>=16 -> K=k0+2,k0+3
// C/D (16x16 f32): VGPR r: lanes0-15 M=r, lanes16-31 M=8+r.
// ---------------------------------------------------------------------------
__global__ void __launch_bounds__(256) proj_gemm(
    const float* __restrict__ H,
    const float* __restrict__ Wq, const float* __restrict__ Wk, const float* __restrict__ Wv,
    const float* __restrict__ bq, const float* __restrict__ bk, const float* __restrict__ bv,
    float* __restrict__ Qo, float* __restrict__ Ko, float* __restrict__ Vo)
{
    const float* W; const float* bias; float* O;
    switch (blockIdx.y) {
        case 0:  W = Wq; bias = bq; O = Qo; break;
        case 1:  W = Wk; bias = bk; O = Ko; break;
        default: W = Wv; bias = bv; O = Vo; break;
    }
    const int wave = threadIdx.x >> 5;
    const int lane = threadIdx.x & 31;
    const int tile = blockIdx.x * 8 + wave;   // 0..4095 (128 Mt x 32 Nt)
    const int m0   = (tile >> 5) * 16;
    const int n0   = (tile & 31) * 16;
    const int mrow = m0 + (lane & 15);
    const int koff = (lane >> 4) * 2;         // 0 or 2
    const int ncol = n0 + (lane & 15);

    v8f acc = {};
    const float* Arow = H + (size_t)mrow * DD + koff;   // 8B-aligned (koff even)
    const float* Bcol = W + (size_t)koff * DD + ncol;

    #pragma unroll 4
    for (int k = 0; k < DD; k += 4) {
        v2f a = *(const v2f*)(Arow + k);                // H[mrow][k+koff .. +1]
        v2f bfr;
        bfr.x = Bcol[(size_t)k * DD];                   // W[k+koff  ][ncol]
        bfr.y = Bcol[(size_t)k * DD + DD];              // W[k+koff+1][ncol]
        acc = __builtin_amdgcn_wmma_f32_16x16x4_f32(
            false, a, false, bfr, (short)0, acc, false, false);
    }

    const float bias_n = bias[ncol];
    float* Orow = O + (size_t)(m0 + (lane >> 4) * 8) * DD + ncol;
    #pragma unroll
    for (int r = 0; r < 8; ++r)
        Orow[(size_t)r * DD] = acc[r] + bias_n;
}

// ---------------------------------------------------------------------------
// Kernel 2: gate = sigmoid(h @ Wg + bg), one wave per row.
// ---------------------------------------------------------------------------
__global__ void __launch_bounds__(256) gate_kernel(
    const float* __restrict__ H, const float* __restrict__ Wg,
    const float* __restrict__ bg, float* __restrict__ G)
{
    const int row  = blockIdx.x * 8 + (threadIdx.x >> 5);
    const int lane = threadIdx.x & 31;
    const float* h = H + (size_t)row * DD;
    float acc = 0.f;
    #pragma unroll 4
    for (int j = lane; j < DD; j += 32) acc += h[j] * Wg[j];
    acc = wave_sum(acc);
    if (lane == 0) {
        float z = acc + bg[0];
        G[row] = 1.f / (1.f + __expf(-z));
    }
}

// ---------------------------------------------------------------------------
// Kernel 3: sequential memory scan. One workgroup per batch element,
// mem (128x512 f32 = 256KB) resident in LDS across all 128 steps.
// ---------------------------------------------------------------------------
#define SM_MEM   0
#define SM_QV    (SS * DD)             // 65536
#define SM_KV    (SM_QV + DD)
#define SM_VV    (SM_KV + DD)
#define SM_RED   (SM_VV + DD)          // 1024 floats
#define SM_DQ    (SM_RED + 1024)       // 128
#define SM_DK    (SM_DQ + SS)
#define SM_RW    (SM_DK + SS)
#define SM_GW    (SM_RW + SS)
#define SM_NS    (SM_GW + SS)
#define SM_SCAL  (SM_NS + SS)          // [0]=nq [1]=nk [2]=gate [3]=mask
#define SM_TOTAL (SM_SCAL + 8)
#define SMEM_BYTES (SM_TOTAL * 4)      // 274,976 B  (< 320KB WGP LDS)

__global__ void __launch_bounds__(1024) scan_kernel(
    const float* __restrict__ init_mem, const float* __restrict__ masks,
    const float* __restrict__ Qraw, const float* __restrict__ Kraw,
    const float* __restrict__ Vraw, const float* __restrict__ Gate,
    float* __restrict__ out)
{
    extern __shared__ float sm[];
    float* mem  = sm + SM_MEM;
    float* qv   = sm + SM_QV;
    float* kv   = sm + SM_KV;
    float* vv   = sm + SM_VV;
    float* red  = sm + SM_RED;
    float* s_dq = sm + SM_DQ;
    float* s_dk = sm + SM_DK;
    float* rw   = sm + SM_RW;
    float* gw   = sm + SM_GW;
    float* s_ns = sm + SM_NS;
    float* scal = sm + SM_SCAL;

    const int b    = blockIdx.x;
    const int tid  = threadIdx.x;
    const int lane = tid & 31;
    const int wave = tid >> 5;

    // Load initial memory state into LDS.
    const float* gm = init_mem + (size_t)b * SS * DD;
    for (int i = tid; i < SS * DD; i += 1024) mem[i] = gm[i];
    __syncthreads();

    for (int t = 0; t < LL; ++t) {
        const int row = b * LL + t;

        // --- Phase A: stage projection rows ---
        if (tid < DD) {
            qv[tid] = Qraw[(size_t)row * DD + tid];
            kv[tid] = Kraw[(size_t)row * DD + tid];
            vv[tid] = Vraw[(size_t)row * DD + tid];
        }
        __syncthreads();

        // norms of q-row (waves 0..15) and k-row (waves 16..31)
        {
            float x = (wave < 16) ? qv[tid] : kv[tid - DD];
            float v = wave_sum(x * x);
            if (lane == 0) red[wave] = v;
        }
        __syncthreads();
        if (tid == 0) {
            float sq = 0.f, sk = 0.f;
            #pragma unroll
            for (int w = 0; w < 16; ++w) { sq += red[w]; sk += red[16 + w]; }
            scal[0] = fmaxf(sqrtf(sq), EPSV);
            scal[1] = fmaxf(sqrtf(sk), EPSV);
            scal[2] = Gate[row];
            scal[3] = masks[(size_t)b * LL + t];
        }
        __syncthreads();

        // --- Phase B: per-slot reductions (dot q, dot k, ||mem_s||^2) ---
        #pragma unroll
        for (int si = 0; si < 4; ++si) {
            const int s = wave * 4 + si;
            const float* mrow = mem + (size_t)s * DD;
            float dq = 0.f, dk = 0.f, ns = 0.f;
            #pragma unroll 4
            for (int j = lane; j < DD; j += 32) {
                float v = mrow[j];
                dq += v * qv[j];
                dk += v * kv[j];
                ns += v * v;
            }
            dq = wave_sum(dq); dk = wave_sum(dk); ns = wave_sum(ns);
            if (lane == 0) { s_dq[s] = dq; s_dk[s] = dk; s_ns[s] = ns; }
        }
        __syncthreads();

        // --- Phase C: two softmaxes (wave 0 -> read weights, wave 1 -> gated write weights)
        if (wave < 2) {
            const float* dd = (wave == 0) ? s_dq : s_dk;
            const float inv_n = 1.f / scal[wave];
            float x[4]; float mx = -1e30f;
            #pragma unroll
            for (int i = 0; i < 4; ++i) {
                const int s = lane + 32 * i;
                float nrm = fmaxf(sqrtf(s_ns[s]), EPSV);
                float sim = dd[s] / nrm * inv_n;
                x[i] = sim * SHARP;
                mx = fmaxf(mx, x[i]);
            }
            mx = wave_max(mx);
            float sum = 0.f;
            #pragma unroll
            for (int i = 0; i < 4; ++i) { x[i] = __expf(x[i] - mx); sum += x[i]; }
            sum = wave_sum(sum);
            const float inv = 1.f / sum;
            if (wave == 0) {
                #pragma unroll
                for (int i = 0; i < 4; ++i) rw[lane + 32 * i] = x[i] * inv;
            } else {
                const float g = scal[2];
                #pragma unroll
                for (int i = 0; i < 4; ++i) gw[lane + 32 * i] = g * x[i] * inv;
            }
        }
        __syncthreads();

        // --- Phase D: fused read_v reduction + gated memory update + mask ---
        {
            const int d    = tid & (DD - 1);
            const int half = tid >> 9;              // 0 or 1 -> slots [0,64) / [64,128)
            const float wvd  = vv[d];
            const float mval = scal[3];
            float acc = 0.f;
            const int s0 = half * 64;
            #pragma unroll 4
            for (int s = s0; s < s0 + 64; ++s) {
                float v = mem[(size_t)s * DD + d];
                acc += rw[s] * v;
                mem[(size_t)s * DD + d] = (v + gw[s] * (wvd - v)) * mval;
            }
            red[tid] = acc;
        }
        __syncthreads();
        if (tid < DD)
            out[(size_t)row * DD + tid] = red[tid] + red[tid + DD];
        __syncthreads();
    }
}

// ---------------------------------------------------------------------------
extern "C" void kernel_launch(void* const* d_in, const int* in_sizes, int n_in,
                              void* d_out, int out_size, void* d_ws, size_t ws_size,
                              hipStream_t stream) {
    const float* init_memory = (const float*)d_in[0];
    const float* hidden      = (const float*)d_in[1];   // (B,L,D) -> rows (B*L, D)
    const float* masks       = (const float*)d_in[2];
    const float* Wq = (const float*)d_in[3];
    const float* bq = (const float*)d_in[4];
    const float* Wk = (const float*)d_in[5];
    const float* bk = (const float*)d_in[6];
    const float* Wv = (const float*)d_in[7];
    const float* bv = (const float*)d_in[8];
    const float* Wg = (const float*)d_in[9];
    const float* bg = (const float*)d_in[10];
    float* out = (float*)d_out;

    float* ws = (float*)d_ws;
    const size_t NROW = (size_t)BB * LL;      // 2048
    float* Q = ws;
    float* K = ws + NROW * DD;
    float* V = ws + 2 * NROW * DD;
    float* G = ws + 3 * NROW * DD;            // ~12.6 MB total

    // Phase 1: hoisted projection GEMMs (WMMA f32) + gate projection.
    proj_gemm<<<dim3(512, 3), 256, 0, stream>>>(hidden, Wq, Wk, Wv, bq, bk, bv, Q, K, V);
    gate_kernel<<<NROW / 8, 256, 0, stream>>>(hidden, Wg, bg, G);

    // Phase 2: sequential scan, memory bank resident in WGP LDS.
    hipFuncSetAttribute((const void*)scan_kernel,
                        hipFuncAttributeMaxDynamicSharedMemorySize, SMEM_BYTES);
    scan_kernel<<<BB, 1024, SMEM_BYTES, stream>>>(init_memory, masks, Q, K, V, G, out);
}